// TrainableMPOLayer_7284264534632
// MI455X (gfx1250) — compile-verified
//
#include <hip/hip_runtime.h>

typedef __attribute__((ext_vector_type(2))) float v2f;
typedef __attribute__((ext_vector_type(4))) float v4f;
typedef __attribute__((ext_vector_type(8))) float v8f;

#define BATCH   2048
#define IN_DIM  4096
#define OUT_DIM 4096
#define ROWS_PER_BLOCK 16
#define THREADS 256

// Fused kernel:
//   s[b,j]   = sum_{k<1024} x[b, j*1024+k],  j = 0..3   (row chunk sums)
//   M[j][m]  = sum_{r1,r2} core0[0,j,k,r1] * cc1[r1,j1,r2] * cc2[r2,j2]
//              with m = k*1024 + j1*32 + j2 (k<4),
//              cc1 = mean(core1, axis=2), cc2 = mean(core2, axis=2)
//   out[b,m] = sum_{j<4} s[b,j] * M[j][m] + bias[m]
// The (2048x4)@(4x4096) GEMM is done with V_WMMA_F32_16X16X4_F32.
__global__ __launch_bounds__(THREADS) void mpo_fused_kernel(
    const float* __restrict__ x,
    const float* __restrict__ core0,
    const float* __restrict__ core1,
    const float* __restrict__ core2,
    const float* __restrict__ bias,
    float* __restrict__ out)
{
    __shared__ float Mlds[4 * OUT_DIM];          // 64 KB: M[j][m]
    __shared__ float cc1s[128];                  // (r1*32+j1)*2+r2
    __shared__ float cc2s[64];                   // r2*32+j2
    __shared__ float sLDS[ROWS_PER_BLOCK * 4];   // s[row][j]

    const int tid  = threadIdx.x;
    const int lane = tid & 31;
    const int wave = tid >> 5;
    const int row_base = blockIdx.x * ROWS_PER_BLOCK;

    // ---- Step A1: collapse core1/core2 over their 3rd axis (means) ----
    if (tid < 128) {
        // flat = r1*64 + j1*2 + r2
        const int r1 = tid >> 6;
        const int j1 = (tid >> 1) & 31;
        const int r2 = tid & 1;
        const float* p = core1 + ((size_t)(r1 * 32 + j1) * 32) * 2 + r2;
        float acc = 0.f;
        #pragma unroll
        for (int q = 0; q < 32; ++q) acc += p[q * 2];
        cc1s[tid] = acc * (1.0f / 32.0f);
    } else if (tid < 192) {
        const int idx = tid - 128;               // r2*32 + j2
        const float* p = core2 + (size_t)idx * 32;
        float acc = 0.f;
        #pragma unroll
        for (int q = 0; q < 32; ++q) acc += p[q];
        cc2s[idx] = acc * (1.0f / 32.0f);
    }
    __syncthreads();

    // ---- Step A2: build M[j][m] in LDS (e = j*4096 + m) ----
    for (int e = tid; e < 4 * OUT_DIM; e += THREADS) {
        const int j  = e >> 12;
        const int m  = e & (OUT_DIM - 1);
        const int k  = m >> 10;
        const int j1 = (m >> 5) & 31;
        const int j2 = m & 31;
        float acc = 0.f;
        #pragma unroll
        for (int r1 = 0; r1 < 2; ++r1) {
            const float c0 = core0[(j * 32 + k) * 2 + r1];
            #pragma unroll
            for (int r2 = 0; r2 < 2; ++r2)
                acc += c0 * cc1s[(r1 * 32 + j1) * 2 + r2] * cc2s[r2 * 32 + j2];
        }
        Mlds[e] = acc;
    }

    // ---- Step B: chunk sums s[row][j]; 64 chunks, 8 per wave ----
    for (int c = wave * 8; c < wave * 8 + 8; ++c) {
        const int row = c >> 2;
        const int jj  = c & 3;
        const v4f* p = (const v4f*)(x + (size_t)(row_base + row) * IN_DIM + jj * 1024);
        v4f a4 = {0.f, 0.f, 0.f, 0.f};
        #pragma unroll
        for (int it = 0; it < 8; ++it)
            a4 += __builtin_nontemporal_load(&p[lane + it * 32]);
        float acc = a4.x + a4.y + a4.z + a4.w;
        #pragma unroll
        for (int off = 16; off > 0; off >>= 1)
            acc += __shfl_xor(acc, off, 32);
        if (lane == 0) sLDS[c] = acc;
    }
    __syncthreads();

    // ---- Step C: out tiles via V_WMMA_F32_16X16X4_F32 ----
    // A (16x4): lane l holds row r=l%16; v0/v1 = K {0,1} (lanes 0-15) or {2,3} (lanes 16-31)
    const int r  = lane & 15;
    const int hi = lane >> 4;
    v2f a;
    a.x = sLDS[r * 4 + hi * 2 + 0];
    a.y = sLDS[r * 4 + hi * 2 + 1];

    for (int t = wave; t < OUT_DIM / 16; t += 8) {
        const int cb = t * 16;
        // B (4x16): lane l holds col n=l%16; v0/v1 = K {0,1} / {2,3}
        v2f b;
        b.x = Mlds[(hi * 2 + 0) * OUT_DIM + cb + r];
        b.y = Mlds[(hi * 2 + 1) * OUT_DIM + cb + r];
        v8f cacc = {};
        v8f d = __builtin_amdgcn_wmma_f32_16x16x4_f32(
            /*neg_a=*/false, a, /*neg_b=*/false, b,
            /*c_mod=*/(short)0, cacc, /*reuse_a=*/false, /*reuse_b=*/false);
        const float bv = bias[cb + r];
        // D: lane l, vgpr v -> row = v + 8*hi, col = r
        #pragma unroll
        for (int v = 0; v < 8; ++v) {
            const int orow = row_base + v + hi * 8;
            __builtin_nontemporal_store(d[v] + bv,
                &out[(size_t)orow * OUT_DIM + cb + r]);
        }
    }
}

extern "C" void kernel_launch(void* const* d_in, const int* in_sizes, int n_in,
                              void* d_out, int out_size, void* d_ws, size_t ws_size,
                              hipStream_t stream) {
    const float* x     = (const float*)d_in[0];
    const float* core0 = (const float*)d_in[1];
    const float* core1 = (const float*)d_in[2];
    const float* core2 = (const float*)d_in[3];
    const float* bias  = (const float*)d_in[4];
    float* out = (float*)d_out;

    dim3 grid(BATCH / ROWS_PER_BLOCK);  // 128 blocks of 16 batch rows
    mpo_fused_kernel<<<grid, THREADS, 0, stream>>>(x, core0, core1, core2, bias, out);
}